// LightGCL_38259568672975
// MI455X (gfx1250) — compile-verified
//
#include <hip/hip_runtime.h>
#include <hip/hip_bf16.h>

#define N_USERS 100000
#define N_ITEMS 50000
#define N_NODES (N_USERS + N_ITEMS)   // 150000
#define EMB 64
#define N_EDGES 1200000
#define N_LAYERS 3

typedef __attribute__((ext_vector_type(2))) float v2f;
typedef __attribute__((ext_vector_type(8))) float v8f;

// ---------------------------------------------------------------------------
// init: x = concat(user_emb, item_emb); acc = x      (float4 vectorized)
// ---------------------------------------------------------------------------
__global__ void lgcn_init(const float* __restrict__ uemb,
                          const float* __restrict__ iemb,
                          float* __restrict__ x,
                          float* __restrict__ acc) {
    const int n4  = N_NODES * EMB / 4;
    const int nu4 = N_USERS * EMB / 4;
    int i = blockIdx.x * blockDim.x + threadIdx.x;
    if (i >= n4) return;
    float4 v = (i < nu4) ? ((const float4*)uemb)[i]
                         : ((const float4*)iemb)[i - nu4];
    ((float4*)x)[i]   = v;
    ((float4*)acc)[i] = v;
}

// ---------------------------------------------------------------------------
// zero: y = 0   (float4 vectorized)
// ---------------------------------------------------------------------------
__global__ void lgcn_zero(float* __restrict__ y) {
    const int n4 = N_NODES * EMB / 4;
    int i = blockIdx.x * blockDim.x + threadIdx.x;
    if (i >= n4) return;
    float4 z; z.x = 0.f; z.y = 0.f; z.z = 0.f; z.w = 0.f;
    ((float4*)y)[i] = z;
}

// ---------------------------------------------------------------------------
// Guaranteed native f32 scatter-add: GLOBAL_ATOMIC_ADD_F32, non-returning
// form (tracked with STOREcnt only, no VGPR writeback, no CAS retry loop).
// Visibility across kernels is provided by kernel-completion (s_endpgm
// implies wait-idle) + stream ordering, so no in-kernel fences are needed.
// ---------------------------------------------------------------------------
__device__ __forceinline__ void atom_add_f32(float* p, float v) {
    asm volatile("global_atomic_add_f32 %0, %1, off"
                 :
                 : "v"(p), "v"(v)
                 : "memory");
}

// ---------------------------------------------------------------------------
// SpMM scatter: y[row[e]] += w[e] * x[col[e]]
// 16 lanes per edge, each lane owns a float4 slice of the 64-dim row.
// Gathers are fully coalesced (16 consecutive lanes read one contiguous
// 256B row as global_load_b128); scatters are native f32 global atomics.
// ---------------------------------------------------------------------------
__global__ void lgcn_spmm(const int*   __restrict__ erow,
                          const int*   __restrict__ ecol,
                          const float* __restrict__ ew,
                          const float* __restrict__ x,
                          float*       __restrict__ y) {
    unsigned tid = blockIdx.x * blockDim.x + threadIdx.x;
    unsigned e = tid >> 4;          // edge index
    unsigned g = tid & 15u;         // dim group: dims [4g, 4g+4)
    if (e >= N_EDGES) return;
    int   r = erow[e];
    int   c = ecol[e];
    float w = ew[e];
    const float4 v = *(const float4*)(x + (size_t)c * EMB + g * 4);
    float* yp = y + (size_t)r * EMB + g * 4;
    atom_add_f32(yp + 0, w * v.x);
    atom_add_f32(yp + 1, w * v.y);
    atom_add_f32(yp + 2, w * v.z);
    atom_add_f32(yp + 3, w * v.w);
}

// ---------------------------------------------------------------------------
// accum: acc += y   (float4 vectorized)
// ---------------------------------------------------------------------------
__global__ void lgcn_accum(float* __restrict__ acc, const float* __restrict__ y) {
    const int n4 = N_NODES * EMB / 4;
    int i = blockIdx.x * blockDim.x + threadIdx.x;
    if (i >= n4) return;
    float4 a = ((float4*)acc)[i];
    float4 b = ((const float4*)y)[i];
    a.x += b.x; a.y += b.y; a.z += b.z; a.w += b.w;
    ((float4*)acc)[i] = a;
}

// ---------------------------------------------------------------------------
// finalize: acc = acc * 0.25, computed with V_WMMA_F32_16X16X4_F32.
// One wave per 16-node block; 4 column tiles of 16; per tile:
//   D = sum_{kk=0..3} (0.25 * P_kk)(16x4) . acc_rows(4x16), C chained.
// P_kk[m][j] = (m == 4kk+j), so D == 0.25 * acc_tile, exactly (pow2 scale,
// RNE exact). Lane layouts per ISA 7.12.2 (A 16x4: K=0,1 lanes 0-15 /
// K=2,3 lanes 16-31; B 4x16 mirrored; C/D: VGPR r -> M=r / M=r+8).
// ---------------------------------------------------------------------------
__global__ void lgcn_finalize_wmma(float* __restrict__ acc) {
    const int lane = threadIdx.x;        // 0..31, one wave32 per block
    const int half = lane >> 4;          // 0 or 1
    const int l    = lane & 15;
    const int k0   = half ? 2 : 0;
    const int k1   = half ? 3 : 1;
    const size_t rowbase = (size_t)blockIdx.x * 16;

#pragma unroll
    for (int t = 0; t < 4; ++t) {        // 4 column tiles of 16 (EMB=64)
        const int ct = t * 16;
        v8f c = {};
#pragma unroll
        for (int kk = 0; kk < 4; ++kk) {
            v2f a, b;
            a.x = (l == 4 * kk + k0) ? 0.25f : 0.0f;   // A[m=l][k0] of 0.25*P_kk
            a.y = (l == 4 * kk + k1) ? 0.25f : 0.0f;   // A[m=l][k1]
            b.x = acc[(rowbase + 4 * kk + k0) * EMB + ct + l];  // B[k0][n=l]
            b.y = acc[(rowbase + 4 * kk + k1) * EMB + ct + l];  // B[k1][n=l]
            c = __builtin_amdgcn_wmma_f32_16x16x4_f32(
                    /*neg_a=*/false, a, /*neg_b=*/false, b,
                    /*c_mod=*/(short)0, c, /*reuse_a=*/false, /*reuse_b=*/false);
        }
#pragma unroll
        for (int r = 0; r < 8; ++r) {
            size_t row = rowbase + (size_t)r + (half ? 8 : 0);
            acc[row * EMB + ct + l] = c[r];
        }
    }
}

// ---------------------------------------------------------------------------
// launch
// ---------------------------------------------------------------------------
extern "C" void kernel_launch(void* const* d_in, const int* in_sizes, int n_in,
                              void* d_out, int out_size, void* d_ws, size_t ws_size,
                              hipStream_t stream) {
    const int*   erow = (const int*)d_in[0];
    const int*   ecol = (const int*)d_in[1];
    const float* ew   = (const float*)d_in[2];
    const float* uemb = (const float*)d_in[3];
    const float* iemb = (const float*)d_in[4];

    float* acc = (float*)d_out;                          // N_NODES*EMB floats
    float* x   = (float*)d_ws;                           // ping
    float* y   = x + (size_t)N_NODES * EMB;              // pong

    const int n4      = N_NODES * EMB / 4;               // 2,400,000
    const int vecBlk  = 256;
    const int vecGrid = (n4 + vecBlk - 1) / vecBlk;      // 9375

    lgcn_init<<<vecGrid, vecBlk, 0, stream>>>(uemb, iemb, x, acc);

    const unsigned spmmThreads = (unsigned)N_EDGES * 16u;     // 19.2M
    const int spmmGrid = (int)((spmmThreads + 255u) / 256u);  // 75000

    for (int layer = 0; layer < N_LAYERS; ++layer) {
        lgcn_zero<<<vecGrid, vecBlk, 0, stream>>>(y);
        lgcn_spmm<<<spmmGrid, 256, 0, stream>>>(erow, ecol, ew, x, y);
        lgcn_accum<<<vecGrid, vecBlk, 0, stream>>>(acc, y);
        float* tmp = x; x = y; y = tmp;                  // pingpong
    }

    // acc holds sum of 4 terms; scale by 0.25 via WMMA (exact pow2 scale).
    lgcn_finalize_wmma<<<N_NODES / 16, 32, 0, stream>>>(acc);
}